// GeneralizedRingAttractorNoGain_1133871366222
// MI455X (gfx1250) — compile-verified
//
#include <hip/hip_runtime.h>
#include <hip/hip_bf16.h>

typedef __attribute__((ext_vector_type(2))) float v2f;
typedef __attribute__((ext_vector_type(8))) float v8f;

#define J0f (-0.1f)
#define J1f (0.1f)
#define ALPHAf (0.15f)
#define Nn 256
#define Bb 64
#define Ss 256
#define Aa 32
#define TWO_PI_OVER_N (6.283185307179586f / 256.0f)

// K dimension of the fused per-step GEMM: (A+1)*N  (last N-block = static W, J0 folded in)
#define KBIG ((Aa + 1) * Nn)        // 8448
#define KCHUNKS (KBIG / 4)          // 2112 fragments of K=4

// ---------------------------------------------------------------------------
// Pack W_big[k=(a,m) or (static,m)][n] into WMMA B-fragment order:
//   addr = ((nt*KCHUNKS + kc)*32 + lane)*2 + v,  k = kc*4 + v + 2*(lane>>4),
//   n = nt*16 + (lane&15)
// k < A*N : Wa[a,n,m] (a=k>>8, m=k&255);  k >= A*N : Wo[n,m]+J1*cos(2pi(n-m)/N)+J0
// ---------------------------------------------------------------------------
__global__ __launch_bounds__(256) void pack_w_kernel(const float* __restrict__ Wa,
                                                     const float* __restrict__ Wo,
                                                     float* __restrict__ Wp) {
  const int kc = blockIdx.x;             // 0..KCHUNKS-1
  const int tid = threadIdx.x;
  const int nt = tid >> 4;
  const int l2 = tid & 15;
  const int n = nt * 16 + l2;
#pragma unroll
  for (int half = 0; half < 2; ++half) {
    const int lane = l2 + half * 16;
#pragma unroll
    for (int v = 0; v < 2; ++v) {
      const int k = kc * 4 + v + half * 2;
      float val;
      if (k < Aa * Nn) {
        const int a = k >> 8, m = k & 255;
        val = Wa[((size_t)a * Nn + n) * Nn + m];
      } else {
        const int m = k - Aa * Nn;
        val = Wo[n * Nn + m] + J1f * cosf(TWO_PI_OVER_N * (float)(n - m)) + J0f;
      }
      Wp[(((size_t)nt * KCHUNKS + kc) * 32 + lane) * 2 + v] = val;
    }
  }
}

// W_delta7[k][n] = cos(2pi(k-n)/N), packed the same way (K=256 -> 64 chunks)
__global__ __launch_bounds__(256) void pack_wd7_kernel(float* __restrict__ Wp) {
  const int kc = blockIdx.x;             // 0..63
  const int tid = threadIdx.x;
  const int nt = tid >> 4;
  const int l2 = tid & 15;
  const int n = nt * 16 + l2;
#pragma unroll
  for (int half = 0; half < 2; ++half) {
    const int lane = l2 + half * 16;
#pragma unroll
    for (int v = 0; v < 2; ++v) {
      const int k = kc * 4 + v + half * 2;
      Wp[((nt * 64 + kc) * 32 + lane) * 2 + v] = cosf(TWO_PI_OVER_N * (float)(n - k));
    }
  }
}

// r0[b,n]: Gaussian bump centered at N/2, circular distance, L2-normalized.
__global__ __launch_bounds__(256) void init_r_kernel(float* __restrict__ r) {
  const int n = threadIdx.x;
  const float d = fabsf((float)n - 128.0f);
  const float dist = fminf(d, 256.0f - d);
  const float wdt = 25.6f;                      // N / 10
  const float bump = expf(-(dist * dist) / (2.0f * wdt * wdt));
  __shared__ float s[256];
  s[n] = bump * bump;
  __syncthreads();
  for (int off = 128; off > 0; off >>= 1) {
    if (n < off) s[n] += s[n + off];
    __syncthreads();
  }
  const float rv = bump / sqrtf(s[0]);
  for (int b = 0; b < Bb; ++b) r[b * Nn + n] = rv;
}

// ---------------------------------------------------------------------------
// One recurrence step: rec = X @ W_big, r_new = (1-a)r + a*relu(rec)
// 64 WGs (4 batch-tiles x 16 n-tiles). Wave w owns a in {4w..4w+3}: one
// ds_load_b64 of r feeds FOUR WMMAs (B fragments at 16KB stride), action
// scalars hoisted to registers. Static W block (multiplier 1) is a separate
// 8-chunk tail per wave. 8 partial tiles LDS-reduced, relu update fused.
// ---------------------------------------------------------------------------
__global__ __launch_bounds__(256) void step_kernel(const float* __restrict__ act,
                                                   const float* __restrict__ Wp,
                                                   const float* __restrict__ r_src,
                                                   float* __restrict__ r_dst,
                                                   float* __restrict__ bump_out,
                                                   int t) {
  __shared__ float r_lds[16 * 256];
  __shared__ float a_lds[16 * 32];
  __shared__ float red[8 * 256];
  const int tid = threadIdx.x;
  const int mt = blockIdx.x >> 4, nt = blockIdx.x & 15;
  const int b0 = mt * 16, n0 = nt * 16;

  for (int e = tid; e < 16 * 256; e += 256)
    r_lds[e] = r_src[(b0 + (e >> 8)) * Nn + (e & 255)];
  for (int e = tid; e < 16 * 32; e += 256)
    a_lds[e] = act[((b0 + (e >> 5)) * Ss + t) * Aa + (e & 31)];
  __syncthreads();

  const int w = tid >> 5, lane = tid & 31;
  const int row = lane & 15;
  const int koff = (lane >> 4) << 1;           // lanes 16..31 hold K+2,K+3
  const float* wbase = Wp + (size_t)nt * ((size_t)KCHUNKS * 64) + lane * 2;
  const float* wdyn = wbase + (size_t)(w * 4 * 64) * 64;   // kc = (4w)*64
  const float* wst  = wbase + (size_t)(2048 + w * 8) * 64; // static tail
  const float* rrow = r_lds + row * 256;

  // hoist the 4 action scalars this wave needs
  const float s0 = a_lds[row * 32 + 4 * w + 0];
  const float s1 = a_lds[row * 32 + 4 * w + 1];
  const float s2 = a_lds[row * 32 + 4 * w + 2];
  const float s3 = a_lds[row * 32 + 4 * w + 3];

  // branch-free prefetch sweep over this wave's B stream (4KB granularity)
#pragma unroll
  for (int p = 0; p < 16; ++p)
    __builtin_prefetch(wdyn + (size_t)p * 1024, 0, 1);

  v8f acc = {};
  for (int mc = 0; mc < 64; ++mc) {
    const v2f rv = *(const v2f*)(rrow + mc * 4 + koff);
    v2f av0, av1, av2, av3;
    av0.x = s0 * rv.x; av0.y = s0 * rv.y;
    av1.x = s1 * rv.x; av1.y = s1 * rv.y;
    av2.x = s2 * rv.x; av2.y = s2 * rv.y;
    av3.x = s3 * rv.x; av3.y = s3 * rv.y;
    const v2f bv0 = *(const v2f*)(wdyn + (size_t)mc * 64 + 0 * 4096);
    const v2f bv1 = *(const v2f*)(wdyn + (size_t)mc * 64 + 1 * 4096);
    const v2f bv2 = *(const v2f*)(wdyn + (size_t)mc * 64 + 2 * 4096);
    const v2f bv3 = *(const v2f*)(wdyn + (size_t)mc * 64 + 3 * 4096);
    acc = __builtin_amdgcn_wmma_f32_16x16x4_f32(false, av0, false, bv0, (short)0, acc,
                                                false, false);
    acc = __builtin_amdgcn_wmma_f32_16x16x4_f32(false, av1, false, bv1, (short)0, acc,
                                                false, false);
    acc = __builtin_amdgcn_wmma_f32_16x16x4_f32(false, av2, false, bv2, (short)0, acc,
                                                false, false);
    acc = __builtin_amdgcn_wmma_f32_16x16x4_f32(false, av3, false, bv3, (short)0, acc,
                                                false, false);
  }
  // static W block: multiplier is 1, A fragment is r directly
#pragma unroll
  for (int mc = 0; mc < 8; ++mc) {
    const int m = (w * 8 + mc) * 4 + koff;
    const v2f rv = *(const v2f*)(rrow + m);
    const v2f bv = *(const v2f*)(wst + (size_t)mc * 64);
    acc = __builtin_amdgcn_wmma_f32_16x16x4_f32(false, rv, false, bv, (short)0, acc,
                                                false, false);
  }

  // reduce the 8 per-wave partial 16x16 tiles
#pragma unroll
  for (int i = 0; i < 8; ++i)
    red[w * 256 + (i + ((lane >> 4) << 3)) * 16 + (lane & 15)] = acc[i];
  __syncthreads();

  float rec = 0.0f;
#pragma unroll
  for (int ww = 0; ww < 8; ++ww) rec += red[ww * 256 + tid];
  const int b = b0 + (tid >> 4), n = n0 + (tid & 15);
  const float rold = r_src[b * Nn + n];
  const float rnew = (1.0f - ALPHAf) * rold + ALPHAf * fmaxf(rec, 0.0f);
  r_dst[b * Nn + n] = rnew;
  bump_out[((size_t)b * Ss + t) * Nn + n] = rnew;
}

// ---------------------------------------------------------------------------
// r_delta7 = bump @ W_delta7 (bump flat = [16384][256]); normalize by row max.
// 1024 WGs: each handles 16 rows x 256 cols; wave w owns n-tiles {2w, 2w+1}.
// ---------------------------------------------------------------------------
__global__ __launch_bounds__(256) void final_kernel(const float* __restrict__ bump,
                                                    const float* __restrict__ Wd7p,
                                                    float* __restrict__ rhist) {
  __shared__ float tile[16 * 256];
  __shared__ float mx[16 * 16];
  __shared__ float rowmax[16];
  const int tid = threadIdx.x;
  const size_t gr0 = (size_t)blockIdx.x * 16;

  for (int e = tid; e < 16 * 256; e += 256) tile[e] = bump[gr0 * 256 + e];
  __syncthreads();

  const int w = tid >> 5, lane = tid & 31;
  const int row = lane & 15;
  const int koff = (lane >> 4) << 1;
  const int nt0 = 2 * w, nt1 = 2 * w + 1;
  const float* b0p = Wd7p + nt0 * 4096 + lane * 2;
  const float* b1p = Wd7p + nt1 * 4096 + lane * 2;
  v8f acc0 = {}, acc1 = {};
  for (int kc = 0; kc < 64; ++kc) {
    const int k = kc << 2;
    const v2f av = *(const v2f*)(tile + row * 256 + k + koff);
    const v2f bv0 = *(const v2f*)(b0p + kc * 64);
    const v2f bv1 = *(const v2f*)(b1p + kc * 64);
    acc0 = __builtin_amdgcn_wmma_f32_16x16x4_f32(false, av, false, bv0, (short)0, acc0,
                                                 false, false);
    acc1 = __builtin_amdgcn_wmma_f32_16x16x4_f32(false, av, false, bv1, (short)0, acc1,
                                                 false, false);
  }
  __syncthreads();  // all waves done reading tile; reuse it for results

#pragma unroll
  for (int i = 0; i < 8; ++i) {
    const int mr = i + ((lane >> 4) << 3);
    tile[mr * 256 + nt0 * 16 + (lane & 15)] = acc0[i];
    tile[mr * 256 + nt1 * 16 + (lane & 15)] = acc1[i];
  }
  __syncthreads();

  {
    const int r2 = tid >> 4, ch = tid & 15;
    float m = tile[r2 * 256 + ch * 16];
#pragma unroll
    for (int j = 1; j < 16; ++j) m = fmaxf(m, tile[r2 * 256 + ch * 16 + j]);
    mx[r2 * 16 + ch] = m;
  }
  __syncthreads();
  if (tid < 16) {
    float m = mx[tid * 16];
    for (int j = 1; j < 16; ++j) m = fmaxf(m, mx[tid * 16 + j]);
    rowmax[tid] = m;
  }
  __syncthreads();

  const int r2 = tid >> 4, c0 = (tid & 15) * 16;
  const float rm = rowmax[r2];
#pragma unroll
  for (int j = 0; j < 16; ++j)
    rhist[(gr0 + r2) * 256 + c0 + j] = tile[r2 * 256 + c0 + j] / rm;
}

// ---------------------------------------------------------------------------
extern "C" void kernel_launch(void* const* d_in, const int* in_sizes, int n_in,
                              void* d_out, int out_size, void* d_ws, size_t ws_size,
                              hipStream_t stream) {
  (void)in_sizes; (void)n_in; (void)out_size;
  const float* act = (const float*)d_in[0];   // [64][256][32]
  const float* Wo  = (const float*)d_in[1];   // [256][256]
  const float* Wa  = (const float*)d_in[2];   // [32][256][256]

  float* out   = (float*)d_out;
  float* rhist = out;                              // [64][256][256]
  float* bump  = out + (size_t)Bb * Ss * Nn;       // [64][256][256]

  float* r0buf = (float*)d_ws;                     // [64][256]
  float* r1buf = r0buf + Bb * Nn;                  // [64][256]
  float* wd7p  = r1buf + Bb * Nn;                  // 256*256 packed
  const size_t small_floats = (size_t)2 * Bb * Nn + (size_t)Nn * Nn;
  const size_t wbig_floats  = (size_t)KBIG * Nn;   // 8448*256 = 8.65 MB
  float* wbig;
  if (ws_size >= (small_floats + wbig_floats) * sizeof(float))
    wbig = wd7p + (size_t)Nn * Nn;                 // lives in workspace
  else
    wbig = rhist;  // stage in r_history output region; final_kernel overwrites it later

  pack_w_kernel<<<KCHUNKS, 256, 0, stream>>>(Wa, Wo, wbig);
  pack_wd7_kernel<<<64, 256, 0, stream>>>(wd7p);
  init_r_kernel<<<1, 256, 0, stream>>>(r0buf);

  for (int t = 0; t < Ss; ++t) {
    const float* rs = (t & 1) ? r1buf : r0buf;
    float*       rd = (t & 1) ? r0buf : r1buf;
    step_kernel<<<64, 256, 0, stream>>>(act, wbig, rs, rd, bump, t);
  }

  final_kernel<<<1024, 256, 0, stream>>>(bump, wd7p, rhist);
}